// _Attention_15522011807752
// MI455X (gfx1250) — compile-verified
//
#include <hip/hip_runtime.h>
#include <hip/hip_bf16.h>

#define B_    8
#define T_    4
#define HID_  2048
#define H_    32
#define HKV_  8
#define D_    64
#define S_    8192
#define SF_   8196          // S + T
#define EPS_  1e-6f
#define SCALE_ 0.125f       // 64^-0.5

typedef float v2f __attribute__((ext_vector_type(2)));
typedef float v8f __attribute__((ext_vector_type(8)));
typedef int   v4i __attribute__((ext_vector_type(4)));

#if defined(__HIP_DEVICE_COMPILE__) && defined(__has_builtin)
#if __has_builtin(__builtin_amdgcn_global_load_async_to_lds_b128) && \
    __has_builtin(__builtin_amdgcn_global_store_async_from_lds_b128)
#define USE_ASYNC_LDS 1
#endif
#endif
#ifndef USE_ASYNC_LDS
#define USE_ASYNC_LDS 0
#endif

#if USE_ASYNC_LDS
// b128 async ops take v4i* in AS1 (global) / AS3 (LDS)
#define GAS128(p) ((__attribute__((address_space(1))) v4i*)(p))
#define LAS128(p) ((__attribute__((address_space(3))) v4i*)(p))
#endif

__device__ __forceinline__ float half_reduce_max(float v) {
    v = fmaxf(v, __shfl_xor(v, 8, 32));
    v = fmaxf(v, __shfl_xor(v, 4, 32));
    v = fmaxf(v, __shfl_xor(v, 2, 32));
    v = fmaxf(v, __shfl_xor(v, 1, 32));
    return v;
}
__device__ __forceinline__ float half_reduce_sum(float v) {
    v += __shfl_xor(v, 8, 32);
    v += __shfl_xor(v, 4, 32);
    v += __shfl_xor(v, 2, 32);
    v += __shfl_xor(v, 1, 32);
    return v;
}

// ---------------------------------------------------------------------------
// Kernel A: QKV projection + per-head RMSNorm + RoPE.
// One 64-thread block per (b, t, head-row). hh<32: q head; 32..39: k; 40..47: v
// ---------------------------------------------------------------------------
__global__ void __launch_bounds__(64)
qkv_kernel(const float* __restrict__ x,
           const float* __restrict__ cs,
           const float* __restrict__ sn,
           const float* __restrict__ wq,
           const float* __restrict__ wk,
           const float* __restrict__ wv,
           const float* __restrict__ qnw,
           const float* __restrict__ knw,
           float* __restrict__ ws_q,
           float* __restrict__ kfull,
           float* __restrict__ vfull) {
    __shared__ float xs[HID_];
    __shared__ float red[D_];
    __shared__ float nv[D_];

    int idx = blockIdx.x;
    int b   = idx / (T_ * 48);
    int rem = idx % (T_ * 48);
    int t   = rem / 48;
    int hh  = rem % 48;
    int tid = threadIdx.x;            // 0..63 (output dim d)

    const float* xrow = x + (size_t)(b * T_ + t) * HID_;
    for (int i = tid; i < HID_; i += 64) xs[i] = xrow[i];
    __syncthreads();

    const float* wmat;
    int ncol, col;
    bool isq = (hh < H_);
    bool isk = (hh >= H_) && (hh < H_ + HKV_);
    if (isq)      { wmat = wq; ncol = H_   * D_; col = hh * D_ + tid; }
    else if (isk) { wmat = wk; ncol = HKV_ * D_; col = (hh - H_) * D_ + tid; }
    else          { wmat = wv; ncol = HKV_ * D_; col = (hh - H_ - HKV_) * D_ + tid; }

    float acc = 0.f;
    const float* wp = wmat + col;
    for (int k = 0; k < HID_; ++k) acc += xs[k] * wp[(size_t)k * ncol];

    if (!isq && !isk) {               // v: no norm, no rope
        int g = hh - 40;
        vfull[(size_t)((b * HKV_ + g) * SF_ + S_ + t) * D_ + tid] = acc;
        return;
    }

    // RMSNorm over the 64-wide head
    red[tid] = acc * acc;
    __syncthreads();
    for (int sred = 32; sred > 0; sred >>= 1) {
        if (tid < sred) red[tid] += red[tid + sred];
        __syncthreads();
    }
    float inv = rsqrtf(red[0] * (1.0f / (float)D_) + EPS_);
    float nrm = (isq ? qnw[tid] : knw[tid]) * acc * inv;

    // RoPE
    nv[tid] = nrm;
    __syncthreads();
    float rot = (tid < 32) ? -nv[tid + 32] : nv[tid - 32];
    float c = cs[(size_t)(b * T_ + t) * D_ + tid];
    float s = sn[(size_t)(b * T_ + t) * D_ + tid];
    float val = nrm * c + rot * s;

    if (isq) {
        int h = hh, g = h >> 2, rep = h & 3;
        ws_q[(size_t)((b * HKV_ + g) * 16 + rep * T_ + t) * D_ + tid] = val;
    } else {
        int g = hh - H_;
        kfull[(size_t)((b * HKV_ + g) * SF_ + S_ + t) * D_ + tid] = val;
    }
}

// ---------------------------------------------------------------------------
// Kernel B: fused flash-decode attention + KV-cache copy.
// One block per (b,g); 16 waves; each wave scans tiles it = w, w+16, ...
// K/V tiles staged via GLOBAL_LOAD_ASYNC_TO_LDS_B128; the cache-append copy
// streams back out of LDS via GLOBAL_STORE_ASYNC_FROM_LDS_B128.
// Score tile and P@V via v_wmma_f32_16x16x4_f32.
// __launch_bounds__(512,1): exact block size; gives the register allocator a
// 1-block/WGP budget so the Q block + O accumulators stay in VGPRs (no spill).
// ---------------------------------------------------------------------------
__global__ void __launch_bounds__(512, 1)
attn_kernel(const float* __restrict__ ck,
            const float* __restrict__ cv,
            const float* __restrict__ ws_q,
            float* __restrict__ kfull,
            float* __restrict__ vfull,
            float* __restrict__ ws_o) {
    extern __shared__ float smem[];
    const int NW = 16;
    int tid = threadIdx.x;
    int w  = tid >> 5;
    int l  = tid & 31;
    int hi = l >> 4;
    int lo = l & 15;
    int bg = blockIdx.x;              // b*HKV + g

    float* kvK = smem + w * 1024;             // [0     .. 16384)
    float* kvV = smem + 16384 + w * 1024;     // [16384 .. 32768)
    float* pb  = smem + 32768 + w * 256;      // [32768 .. 36864)
    float* sm  = smem + 36864;                // 256
    float* sl  = smem + 37120;                // 256
    float* Mg  = smem + 37376;                // 16
    float* Gg  = smem + 37392;                // 16
    float* Oc  = smem + 37408;                // 1024  -> total 38432 floats

    for (int e = tid; e < 1024; e += 512) Oc[e] = 0.f;
    __syncthreads();

    // Q block (16 rows x 64) in native WMMA A-layout registers
    const float* qb = ws_q + (size_t)bg * 16 * D_ + (size_t)lo * D_;
    v2f qa[16];
#pragma unroll
    for (int kk = 0; kk < 16; ++kk)
        qa[kk] = *(const v2f*)(qb + 4 * kk + 2 * hi);

    float run_m[8], run_l[8];
#pragma unroll
    for (int r = 0; r < 8; ++r) { run_m[r] = -1e30f; run_l[r] = 0.f; }
    v8f oacc[4];
#pragma unroll
    for (int nt = 0; nt < 4; ++nt)
#pragma unroll
        for (int r = 0; r < 8; ++r) oacc[nt][r] = 0.f;

    const size_t cbase = (size_t)bg * S_  * D_;
    const size_t fbase = (size_t)bg * SF_ * D_;

    for (int it = w; it < 513; it += NW) {
        int s0 = it * 16;
        // prior async copy-stores read kvK/kvV from LDS; drain before reuse
        asm volatile("s_wait_asynccnt 0x0" ::: "memory");

        if (it + NW < 512) {
            __builtin_prefetch(ck + cbase + (size_t)(s0 + NW * 16) * D_, 0, 0);
            __builtin_prefetch(cv + cbase + (size_t)(s0 + NW * 16) * D_, 0, 0);
        }

        // ---- stage K & V tiles into LDS ----
        if (it < 512) {
#if USE_ASYNC_LDS
            const float* gk = ck + cbase + (size_t)s0 * D_ + (size_t)l * 4;
            const float* gv = cv + cbase + (size_t)s0 * D_ + (size_t)l * 4;
            float* lk = kvK + l * 4;
            float* lv = kvV + l * 4;
#pragma unroll
            for (int j = 0; j < 8; ++j) {
                __builtin_amdgcn_global_load_async_to_lds_b128(
                    GAS128(gk + j * 128), LAS128(lk + j * 128), 0, 0);
                __builtin_amdgcn_global_load_async_to_lds_b128(
                    GAS128(gv + j * 128), LAS128(lv + j * 128), 0, 0);
            }
            asm volatile("s_wait_asynccnt 0x0" ::: "memory");
#else
            const float4* srck = (const float4*)(ck + cbase + (size_t)s0 * D_);
            const float4* srcv = (const float4*)(cv + cbase + (size_t)s0 * D_);
            float4* dstk = (float4*)(kfull + fbase + (size_t)s0 * D_);
            float4* dstv = (float4*)(vfull + fbase + (size_t)s0 * D_);
            float4* ldsk = (float4*)kvK;
            float4* ldsv = (float4*)kvV;
#pragma unroll
            for (int j = 0; j < 8; ++j) {
                float4 vk = srck[j * 32 + l];
                float4 vv = srcv[j * 32 + l];
                ldsk[j * 32 + l] = vk;  dstk[j * 32 + l] = vk;
                ldsv[j * 32 + l] = vv;  dstv[j * 32 + l] = vv;
            }
            asm volatile("s_wait_dscnt 0x0" ::: "memory");
#endif
        } else {                      // tail tile: 4 new tokens (from qkv), rest 0
            const float4* srck = (const float4*)(kfull + fbase + (size_t)s0 * D_);
            const float4* srcv = (const float4*)(vfull + fbase + (size_t)s0 * D_);
            float4* ldsk = (float4*)kvK;
            float4* ldsv = (float4*)kvV;
#pragma unroll
            for (int j = 0; j < 8; ++j) {
                int fi = j * 32 + l;
                float4 vk = make_float4(0.f, 0.f, 0.f, 0.f);
                float4 vv = make_float4(0.f, 0.f, 0.f, 0.f);
                if (s0 + (fi >> 2) < SF_) { vk = srck[fi]; vv = srcv[fi]; }
                ldsk[fi] = vk;
                ldsv[fi] = vv;
            }
            asm volatile("s_wait_dscnt 0x0" ::: "memory");
        }

        // ---- 16x16 score tile: 16 x v_wmma_f32_16x16x4_f32 ----
        v8f sc;
#pragma unroll
        for (int r = 0; r < 8; ++r) sc[r] = 0.f;
#pragma unroll
        for (int kk = 0; kk < 16; ++kk) {
            v2f bk = *(const v2f*)(kvK + lo * D_ + 4 * kk + 2 * hi);
            sc = __builtin_amdgcn_wmma_f32_16x16x4_f32(
                    false, qa[kk], false, bk, (short)0, sc, false, false);
        }

        // ---- online softmax (row = r + 8*hi lives in a 16-lane half) ----
        bool valid = (s0 + lo) < SF_;
        float p[8];
#pragma unroll
        for (int r = 0; r < 8; ++r) {
            float sv   = valid ? sc[r] * SCALE_ : -1e30f;
            float tmax = half_reduce_max(sv);
            float nm    = fmaxf(run_m[r], tmax);
            float alpha = __expf(run_m[r] - nm);
            float pe    = valid ? __expf(sv - nm) : 0.f;
            float tsum  = half_reduce_sum(pe);
            run_l[r] = run_l[r] * alpha + tsum;
            run_m[r] = nm;
            p[r] = pe;
#pragma unroll
            for (int nt = 0; nt < 4; ++nt) oacc[nt][r] *= alpha;
        }

        // ---- P tile to LDS (C-layout -> A-layout round trip) ----
#pragma unroll
        for (int r = 0; r < 8; ++r) pb[(r + 8 * hi) * 16 + lo] = p[r];
        asm volatile("s_wait_dscnt 0x0" ::: "memory");

        // ---- O += P @ V : 4 n-tiles x 4 k-steps of wmma ----
#pragma unroll
        for (int nt = 0; nt < 4; ++nt) {
#pragma unroll
            for (int kk = 0; kk < 4; ++kk) {
                v2f pa = *(const v2f*)(pb + lo * 16 + 4 * kk + 2 * hi);
                v2f bv;
                bv[0] = kvV[(4 * kk + 2 * hi) * D_ + nt * 16 + lo];
                bv[1] = kvV[(4 * kk + 2 * hi + 1) * D_ + nt * 16 + lo];
                oacc[nt] = __builtin_amdgcn_wmma_f32_16x16x4_f32(
                        false, pa, false, bv, (short)0, oacc[nt], false, false);
            }
        }

#if USE_ASYNC_LDS
        // ---- cache-append copy streamed straight from LDS ----
        if (it < 512) {
            float* dk = kfull + fbase + (size_t)s0 * D_ + (size_t)l * 4;
            float* dv = vfull + fbase + (size_t)s0 * D_ + (size_t)l * 4;
            const float* lk = kvK + l * 4;
            const float* lv = kvV + l * 4;
#pragma unroll
            for (int j = 0; j < 8; ++j) {
                __builtin_amdgcn_global_store_async_from_lds_b128(
                    GAS128(dk + j * 128), LAS128(lk + j * 128), 0, 0);
                __builtin_amdgcn_global_store_async_from_lds_b128(
                    GAS128(dv + j * 128), LAS128(lv + j * 128), 0, 0);
            }
        }
#endif
    }
    asm volatile("s_wait_asynccnt 0x0" ::: "memory");

    // ---- cross-wave combine ----
    if (lo == 0) {
#pragma unroll
        for (int r = 0; r < 8; ++r) {
            int row = r + 8 * hi;
            sm[w * 16 + row] = run_m[r];
            sl[w * 16 + row] = run_l[r];
        }
    }
    __syncthreads();
    if (tid < 16) {
        float M = -1e30f;
        for (int w2 = 0; w2 < NW; ++w2) M = fmaxf(M, sm[w2 * 16 + tid]);
        float G = 0.f;
        for (int w2 = 0; w2 < NW; ++w2) G += sl[w2 * 16 + tid] * __expf(sm[w2 * 16 + tid] - M);
        Mg[tid] = M;
        Gg[tid] = G;
    }
    __syncthreads();
#pragma unroll
    for (int r = 0; r < 8; ++r) {
        int row = r + 8 * hi;
        float scl = __expf(run_m[r] - Mg[row]);
#pragma unroll
        for (int nt = 0; nt < 4; ++nt)
            atomicAdd(&Oc[row * D_ + nt * 16 + lo], oacc[nt][r] * scl);
    }
    __syncthreads();
    for (int e = tid; e < 1024; e += 512) {
        int row = e >> 6;
        ws_o[(size_t)bg * 1024 + e] = Oc[e] / Gg[row];
    }
}

// ---------------------------------------------------------------------------
// Kernel C: output projection. One block per (b,t); permuted og row staged
// in LDS, coalesced dot against wo columns.
// ---------------------------------------------------------------------------
__global__ void __launch_bounds__(256)
out_kernel(const float* __restrict__ ws_o,
           const float* __restrict__ wo,
           float* __restrict__ out) {
    __shared__ float vec[HID_];
    int b = blockIdx.x / T_;
    int t = blockIdx.x % T_;
    int tid = threadIdx.x;            // 256 threads

    for (int e = tid; e < HID_; e += 256) {
        int h = e >> 6, d = e & 63;
        int g = h >> 2, rep = h & 3;
        vec[e] = ws_o[(size_t)((b * HKV_ + g) * 16 + rep * T_ + t) * D_ + d];
    }
    __syncthreads();

#pragma unroll
    for (int j = 0; j < 8; ++j) {
        int c = tid + j * 256;
        float acc = 0.f;
        for (int k = 0; k < HID_; ++k) acc += vec[k] * wo[(size_t)k * HID_ + c];
        out[(size_t)(b * T_ + t) * HID_ + c] = acc;
    }
}

extern "C" void kernel_launch(void* const* d_in, const int* in_sizes, int n_in,
                              void* d_out, int out_size, void* d_ws, size_t ws_size,
                              hipStream_t stream) {
    const float* x   = (const float*)d_in[0];
    const float* cs  = (const float*)d_in[1];
    const float* sn  = (const float*)d_in[2];
    const float* ck  = (const float*)d_in[3];
    const float* cv  = (const float*)d_in[4];
    const float* wq  = (const float*)d_in[5];
    const float* wk  = (const float*)d_in[6];
    const float* wv  = (const float*)d_in[7];
    const float* wo  = (const float*)d_in[8];
    const float* qnw = (const float*)d_in[9];
    const float* knw = (const float*)d_in[10];

    float* out   = (float*)d_out;                          // (8,4,2048)
    float* kfull = out + (size_t)B_ * T_ * HID_;           // (8,8,8196,64)
    float* vfull = kfull + (size_t)B_ * HKV_ * SF_ * D_;   // (8,8,8196,64)

    float* ws   = (float*)d_ws;
    float* ws_q = ws;                                      // 65536 floats
    float* ws_o = ws + 65536;                              // 65536 floats

    qkv_kernel<<<B_ * T_ * 48, 64, 0, stream>>>(x, cs, sn, wq, wk, wv, qnw, knw,
                                                ws_q, kfull, vfull);
    attn_kernel<<<B_ * HKV_, 512, 38432 * sizeof(float), stream>>>(ck, cv, ws_q,
                                                                   kfull, vfull, ws_o);
    out_kernel<<<B_ * T_, 256, 0, stream>>>(ws_o, wo, out);
}